// GRN_2473901163257
// MI455X (gfx1250) — compile-verified
//
#include <hip/hip_runtime.h>
#include <hip/hip_bf16.h>

#define N_NODES 50000
#define N_EDGES 800000
#define NF 64          // IN_NF = HID = OUT_NF = 64
#define EDGE_K 132     // 2*64 + 3 radial, padded to multiple of 4

typedef float v2f __attribute__((ext_vector_type(2)));
typedef float v8f __attribute__((ext_vector_type(8)));

// ---------------------------------------------------------------------------
// Stage a [K][64] row-major weight matrix into WMMA-B fragment order in LDS.
// For V_WMMA_F32_16X16X4_F32, B is 4x16 per n-tile; 32-bit B striping:
//   lane = 16*(k%4>=2) + (n%16), vgpr j = (k%4)&1, rows K..K+3 per chunk.
// dst layout: [(kc*4 + ntile)*64 + lane*2 + j], zero-padded (K=131 -> 132).
// ---------------------------------------------------------------------------
__device__ __forceinline__ void stage_frag(const float* __restrict__ W, int K,
                                           float* dst, int kChunks,
                                           int tid, int nthr) {
  int total = kChunks * 256;
  for (int i = tid; i < total; i += nthr) dst[i] = 0.0f;
  __syncthreads();
  int tot2 = K * 64;
  for (int idx = tid; idx < tot2; idx += nthr) {
    int k = idx >> 6, n = idx & 63;
    int kc = k >> 2, r = k & 3;
    int lane = ((r >> 1) << 4) | (n & 15);
    int nt = n >> 4;
    dst[((kc * 4 + nt) << 6) + (lane << 1) + (r & 1)] = W[idx];
  }
}

// ---------------------------------------------------------------------------
// 16-row MLP layer on the fp32 WMMA pipe: D[16x64] = act(A[16xK] @ W + b).
// A rows in (LDS) memory at stride rs; W pre-staged in fragment order.
// A fragment (32-bit A 16x4): lane m=L&15, VGPR0/1 = K{0,1} (lo half) or
// K{2,3} (hi half) of each 4-wide chunk.  D: vgpr r -> row r + 8*(L>=16),
// col (L&15) + 16*ntile.
// ---------------------------------------------------------------------------
__device__ __forceinline__ void mlp16(const float* aBase, int rs, int kChunks,
                                      const float* wfrag, const float* bias,
                                      float* dBase, int dRs, bool doRelu,
                                      int lane) {
  const int m = lane & 15;
  const int hi = lane >> 4;
  v8f acc[4];
  const v8f zero = {0.f, 0.f, 0.f, 0.f, 0.f, 0.f, 0.f, 0.f};
#pragma unroll
  for (int nt = 0; nt < 4; ++nt) acc[nt] = zero;

  const float* aRow = aBase + m * rs;
  for (int kc = 0; kc < kChunks; ++kc) {
    int kb = (kc << 2) + (hi << 1);
    v2f a;
    a.x = aRow[kb];
    a.y = aRow[kb + 1];
    const float* wp = wfrag + (kc << 8) + (lane << 1);
#pragma unroll
    for (int nt = 0; nt < 4; ++nt) {
      v2f b;
      b.x = wp[nt << 6];
      b.y = wp[(nt << 6) + 1];
      // D = A*B + C, fp32 matrix pipe (V_WMMA_F32_16X16X4_F32)
      acc[nt] = __builtin_amdgcn_wmma_f32_16x16x4_f32(
          false, a, false, b, (short)0, acc[nt], false, false);
    }
  }
#pragma unroll
  for (int nt = 0; nt < 4; ++nt) {
    int n = m + (nt << 4);
    float bv = bias[n];
    float* drow = dBase + n;
#pragma unroll
    for (int r = 0; r < 8; ++r) {
      float v = acc[nt][r] + bv;
      if (doRelu) v = fmaxf(v, 0.0f);
      drow[(r + (hi << 3)) * dRs] = v;
    }
  }
}

// ---------------------------------------------------------------------------
// Edge kernel: per-wave 16-edge tiles.
//   edge_in = [h[row] | h[col] | fx fv fxv 0]  (16 x 132 in LDS)
//   m1 = relu(edge_in@W1+b1) -> sx[:,0:64]
//   m2 = relu(m1@W2+b2)      -> sx[:,64:128]; atomic scatter to agg[row]
//   p  = relu(m2@C1+cb1)     -> sx[:,0:64]
//   s  = p @ c_w; trans = clip(coord_diff*s); scatter to seg[row], cnt[row]
// ---------------------------------------------------------------------------
__global__ void __launch_bounds__(256) egnn_edge(
    const float* __restrict__ h, const float* __restrict__ coord,
    const float* __restrict__ vel, const int* __restrict__ ei,
    const float* __restrict__ eW1, const float* __restrict__ eB1,
    const float* __restrict__ eW2, const float* __restrict__ eB2,
    const float* __restrict__ cW1, const float* __restrict__ cB1,
    const float* __restrict__ cW,
    float* __restrict__ agg, float* __restrict__ seg, float* __restrict__ cnt) {
  __shared__ float sW1f[33 * 256];   // 131->132 padded, 33 k-chunks
  __shared__ float sW2f[16 * 256];
  __shared__ float sC1f[16 * 256];
  __shared__ float sB1[64], sB2[64], sCB1[64], sCWv[64];
  __shared__ float sX[8][16 * EDGE_K];
  __shared__ float sCD[8][16][4];
  __shared__ int sRowI[8][16];
  __shared__ int sColI[8][16];

  const int tid = threadIdx.x, nthr = blockDim.x;
  stage_frag(eW1, 131, sW1f, 33, tid, nthr);
  stage_frag(eW2, 64, sW2f, 16, tid, nthr);
  stage_frag(cW1, 64, sC1f, 16, tid, nthr);
  if (tid < 64) {
    sB1[tid] = eB1[tid];
    sB2[tid] = eB2[tid];
    sCB1[tid] = cB1[tid];
    sCWv[tid] = cW[tid];
  }
  __syncthreads();

  const int wave = tid >> 5;
  const int lane = tid & 31;
  float* sx = sX[wave];
  const int half = lane >> 4;
  const int l4 = (lane & 15) << 2;
  const int nTiles = N_EDGES / 16;

  for (int tile = blockIdx.x * 8 + wave; tile < nTiles; tile += gridDim.x * 8) {
    const int e0 = tile * 16;
    // radial features + coord_diff (lanes 0-15, one edge each)
    if (lane < 16) {
      int e = e0 + lane;
      int r_ = ei[e];
      int c_ = ei[N_EDGES + e];
      sRowI[wave][lane] = r_;
      sColI[wave][lane] = c_;
      float cd0 = coord[r_ * 3 + 0] - coord[c_ * 3 + 0];
      float cd1 = coord[r_ * 3 + 1] - coord[c_ * 3 + 1];
      float cd2 = coord[r_ * 3 + 2] - coord[c_ * 3 + 2];
      float vd0 = vel[r_ * 3 + 0] - vel[c_ * 3 + 0];
      float vd1 = vel[r_ * 3 + 1] - vel[c_ * 3 + 1];
      float vd2 = vel[r_ * 3 + 2] - vel[c_ * 3 + 2];
      float fx = sqrtf(cd0 * cd0 + cd1 * cd1 + cd2 * cd2);
      float fv = sqrtf(vd0 * vd0 + vd1 * vd1 + vd2 * vd2);
      float fxv = (cd0 * vd0 + cd1 * vd1 + cd2 * vd2) / (fx * fv);
      float* xr = sx + lane * EDGE_K;
      xr[128] = fx;
      xr[129] = fv;
      xr[130] = fxv;
      xr[131] = 0.0f;
      sCD[wave][lane][0] = cd0;
      sCD[wave][lane][1] = cd1;
      sCD[wave][lane][2] = cd2;
    }
    // gather h[row] (lanes 0-15) / h[col] (lanes 16-31): one b128 per edge/lane
    for (int e = 0; e < 16; ++e) {
      int node = half ? sColI[wave][e] : sRowI[wave][e];
      float4 v = ((const float4*)(h + node * 64))[l4 >> 2];
      *(float4*)(sx + e * EDGE_K + half * 64 + l4) = v;
    }
    // edge MLP on the fp32 WMMA pipe
    mlp16(sx, EDGE_K, 33, sW1f, sB1, sx, EDGE_K, true, lane);       // m1
    mlp16(sx, EDGE_K, 16, sW2f, sB2, sx + 64, EDGE_K, true, lane);  // m2
    // scatter m2 into agg[row] (segment_sum for node model)
    for (int e = 0; e < 16; ++e) {
      int r_ = sRowI[wave][e];
      float v0 = sx[e * EDGE_K + 64 + lane];
      float v1 = sx[e * EDGE_K + 96 + lane];
      atomicAdd(&agg[r_ * 64 + lane], v0);
      atomicAdd(&agg[r_ * 64 + 32 + lane], v1);
    }
    // coord head
    mlp16(sx + 64, EDGE_K, 16, sC1f, sCB1, sx, EDGE_K, true, lane);  // p
    if (lane < 16) {
      const float* pr = sx + lane * EDGE_K;
      float s = 0.0f;
      for (int n2 = 0; n2 < 64; ++n2) s += pr[n2] * sCWv[n2];
      int r_ = sRowI[wave][lane];
#pragma unroll
      for (int d = 0; d < 3; ++d) {
        float t = sCD[wave][lane][d] * s;
        t = fminf(fmaxf(t, -100.0f), 100.0f);
        atomicAdd(&seg[r_ * 3 + d], t);
      }
      atomicAdd(&cnt[r_], 1.0f);
    }
  }
}

// ---------------------------------------------------------------------------
// Node kernel: per-wave 16-node tiles.
//   x = [h | agg] (16 x 128); y = relu(x@nW1+b1); h_out = y@nW2+b2 (global)
//   coord_out = seg / max(cnt,1)
// ---------------------------------------------------------------------------
__global__ void __launch_bounds__(256) egnn_node(
    const float* __restrict__ h,
    const float* __restrict__ nW1, const float* __restrict__ nB1,
    const float* __restrict__ nW2, const float* __restrict__ nB2,
    const float* __restrict__ agg, const float* __restrict__ seg,
    const float* __restrict__ cnt,
    float* __restrict__ hout, float* __restrict__ cout) {
  __shared__ float sW1f[32 * 256];  // K=128 -> 32 chunks
  __shared__ float sW2f[16 * 256];
  __shared__ float sB1[64], sB2[64];
  __shared__ float sX[8][16 * 128];

  const int tid = threadIdx.x, nthr = blockDim.x;
  stage_frag(nW1, 128, sW1f, 32, tid, nthr);
  stage_frag(nW2, 64, sW2f, 16, tid, nthr);
  if (tid < 64) {
    sB1[tid] = nB1[tid];
    sB2[tid] = nB2[tid];
  }
  __syncthreads();

  const int wave = tid >> 5;
  const int lane = tid & 31;
  float* sx = sX[wave];
  const int half = lane >> 4;
  const int l4 = (lane & 15) << 2;
  const int nTiles = N_NODES / 16;

  for (int tile = blockIdx.x * 8 + wave; tile < nTiles; tile += gridDim.x * 8) {
    const int base = tile * 16;
    for (int e = 0; e < 16; ++e) {
      const float* src = half ? (agg + (base + e) * 64) : (h + (base + e) * 64);
      float4 v = ((const float4*)src)[l4 >> 2];
      *(float4*)(sx + e * 128 + half * 64 + l4) = v;
    }
    mlp16(sx, 128, 32, sW1f, sB1, sx, 128, true, lane);            // hidden
    mlp16(sx, 128, 16, sW2f, sB2, hout + base * 64, 64, false, lane);  // h_out
    if (lane < 16) {
      int node = base + lane;
      float c = fmaxf(cnt[node], 1.0f);
#pragma unroll
      for (int d = 0; d < 3; ++d)
        cout[node * 3 + d] = seg[node * 3 + d] / c;
    }
  }
}

extern "C" void kernel_launch(void* const* d_in, const int* in_sizes, int n_in,
                              void* d_out, int out_size, void* d_ws,
                              size_t ws_size, hipStream_t stream) {
  const float* h     = (const float*)d_in[0];
  const float* coord = (const float*)d_in[1];
  const float* vel   = (const float*)d_in[2];
  const int*   ei    = (const int*)d_in[3];
  const float* eW1 = (const float*)d_in[4];
  const float* eB1 = (const float*)d_in[5];
  const float* eW2 = (const float*)d_in[6];
  const float* eB2 = (const float*)d_in[7];
  const float* nW1 = (const float*)d_in[8];
  const float* nB1 = (const float*)d_in[9];
  const float* nW2 = (const float*)d_in[10];
  const float* nB2 = (const float*)d_in[11];
  const float* cW1 = (const float*)d_in[12];
  const float* cB1 = (const float*)d_in[13];
  const float* cW  = (const float*)d_in[14];

  float* ws = (float*)d_ws;
  float* agg = ws;                       // [N,64]
  float* seg = ws + (size_t)N_NODES * 64;  // [N,3]
  float* cnt = ws + (size_t)N_NODES * 67;  // [N]
  hipMemsetAsync(d_ws, 0, (size_t)N_NODES * 68 * sizeof(float), stream);

  egnn_edge<<<625, 256, 0, stream>>>(h, coord, vel, ei, eW1, eB1, eW2, eB2,
                                     cW1, cB1, cW, agg, seg, cnt);

  float* hout = (float*)d_out;
  float* cout = hout + (size_t)N_NODES * 64;
  egnn_node<<<391, 256, 0, stream>>>(h, nW1, nB1, nW2, nB2, agg, seg, cnt,
                                     hout, cout);
}